// FillPolygon_58720792871700
// MI455X (gfx1250) — compile-verified
//
#include <hip/hip_runtime.h>
#include <hip/hip_bf16.h>
#include <math.h>

// FillPolygon on MI455X (gfx1250):
//  - tiny, exp/VALU-bound problem (1 MB out @ 23.3 TB/s ~ 45 ns; ~100M v_exp_f32)
//  - stage tt1[384]/r1[384] in LDS (broadcast reads), e computed in registers,
//  - accumulation via V_WMMA_F32_16X16X4_F32: D = [ones; r1] x E, rows 0/1 of the
//    16x16 f32 accumulator are the denominator / numerator for 16 pixels per wave.
//  - exp(-100*min(d,1)) folded to native exp2: exp2(max(d*S, S)), S = -100*log2(e),
//    saving one packed multiply per e-pair vs __expf.

typedef float v2f __attribute__((ext_vector_type(2)));
typedef float v8f __attribute__((ext_vector_type(8)));

#define TWO_PI_F   6.2831853071795864769f
#define EXP2_SCALE (-144.26950408889634074f)   // -100 / ln(2)

__device__ __forceinline__ float fix_radians(float t) {
    // exact rewrite of: -relu(-(2pi - relu(-t))) + relu(t)
    float a = fmaxf(-t, 0.0f);
    float b = fminf(TWO_PI_F - a, 0.0f);
    return b + fmaxf(t, 0.0f);
}

__global__ __launch_bounds__(256) void fill_polygon_wmma(
    const float* __restrict__ in,   // (B=4, 128, 2) f32
    float* __restrict__ out)        // (B=4, 256, 256) f32
{
    __shared__ float s_tt[384];     // tt1 = [tt-2pi | tt | tt+2pi]
    __shared__ float s_r[384];      // r1  = [r | r | r]

    const int tid  = threadIdx.x;
    const int b    = blockIdx.x >> 9;              // 512 blocks per batch
    const int base = (blockIdx.x & 511) * 128;     // 128 pixels per block

    // ---- stage per-batch point data into LDS ----
    if (tid < 128) {
        float x  = in[(b * 128 + tid) * 2 + 0];
        float y  = in[(b * 128 + tid) * 2 + 1];
        float dx = x - 128.0f;
        float dy = y - 128.0f;
        float r  = sqrtf(dx * dx + dy * dy);
        float tt = fix_radians(atan2f(dy, dx));
        s_tt[tid]       = tt - TWO_PI_F;
        s_tt[128 + tid] = tt;
        s_tt[256 + tid] = tt + TWO_PI_F;
        s_r[tid]        = r;
        s_r[128 + tid]  = r;
        s_r[256 + tid]  = r;
    }
    __syncthreads();

    // ---- per-lane geometry: wave owns 16 pixels (columns), lanes 16-31 do K=2,3 ----
    const int lane = tid & 31;
    const int wave = tid >> 5;
    const int col  = lane & 15;             // pixel column within wave tile
    const int koff = (lane >> 4) << 1;      // 0 for lanes 0-15, 2 for lanes 16-31

    const int   q   = base + wave * 16 + col;          // flat pixel in batch
    const float gx  = (float)(q >> 8)  - 128.0f;       // row index - HALF_X
    const float gy  = (float)(q & 255) - 128.0f;       // col index - HALF_Y
    const float rm  = sqrtf(gx * gx + gy * gy);
    const float ttp = fix_radians(atan2f(gy, gx));

    // A-matrix row masks: row 0 = ones (denominator), row 1 = r1 (numerator)
    const float m0 = (col == 0) ? 1.0f : 0.0f;
    const float m1 = (col == 1) ? 1.0f : 0.0f;

    v8f acc = {0.0f, 0.0f, 0.0f, 0.0f, 0.0f, 0.0f, 0.0f, 0.0f};

    #pragma unroll 8
    for (int jb = 0; jb < 384; jb += 4) {
        const int idx = jb + koff;          // this lane's K-pair of point indices
        float t0  = s_tt[idx];
        float t1  = s_tt[idx + 1];
        float rv0 = s_r[idx];
        float rv1 = s_r[idx + 1];

        // e = exp(-100 * min(d,1)) == exp2(max(d*S, S)), S = -100*log2(e)
        float d0 = t0 - ttp; d0 = d0 * d0;
        float d1 = t1 - ttp; d1 = d1 * d1;
        float a0 = fmaxf(d0 * EXP2_SCALE, EXP2_SCALE);
        float a1 = fmaxf(d1 * EXP2_SCALE, EXP2_SCALE);
        float e0 = __builtin_amdgcn_exp2f(a0);   // native v_exp_f32 (TRANS pipe)
        float e1 = __builtin_amdgcn_exp2f(a1);

        v2f A;                               // A 16x4 f32: VGPR0=K{0|2}, VGPR1=K{1|3}
        A.x = fmaf(m1, rv0, m0);
        A.y = fmaf(m1, rv1, m0);
        v2f Bm;                              // B 4x16 f32: e-values for this column
        Bm.x = e0;
        Bm.y = e1;

        // D = A x B + C  (16x16x4 f32)
        acc = __builtin_amdgcn_wmma_f32_16x16x4_f32(
                  false, A, false, Bm, (short)0, acc, false, false);
    }

    // row 0 (den) and row 1 (num) of D live in VGPR0/VGPR1 of lanes 0-15
    if (lane < 16) {
        float den  = acc[0];
        float num  = acc[1];
        float flag = fmaxf(1.0f + num / den - rm, 0.0f);
        out[b * 65536 + q] = flag;
    }
}

extern "C" void kernel_launch(void* const* d_in, const int* in_sizes, int n_in,
                              void* d_out, int out_size, void* d_ws, size_t ws_size,
                              hipStream_t stream) {
    (void)in_sizes; (void)n_in; (void)d_ws; (void)ws_size; (void)out_size;
    const float* in = (const float*)d_in[0];
    float* out = (float*)d_out;
    // 4 batches * 512 blocks, 256 threads (8 waves of 16 pixels each)
    fill_polygon_wmma<<<dim3(2048), dim3(256), 0, stream>>>(in, out);
}